// ExtendSkeletonGraphConv_52630529245312
// MI455X (gfx1250) — compile-verified
//
#include <hip/hip_runtime.h>

// ---------------------------------------------------------------------------
// ExtendSkeletonGraphConv fused kernel for MI455X (gfx1250, wave32, WMMA)
//
// out[b,n,f] = D[n,f]*h0[b,n,f] + sum_j T[n,j,f]*h1[b,j,f] + bias[f]
//   h0 = x @ W0, h1 = x @ W1
//   D,T precomputed from adjacencies + Lamdas (batch independent, tiny)
// ---------------------------------------------------------------------------

typedef __attribute__((ext_vector_type(2))) float v2f;
typedef __attribute__((ext_vector_type(8))) float v8f;

#define NJ      17
#define FDIM    128
#define TB      16              // batches per workgroup
#define ROWS    (TB * NJ)       // 272 rows (17 WMMA m-tiles)
#define THREADS 256             // 8 waves

// ---------------------------------------------------------------------------
// Precompute D[n,f] and T[n,j,f] from adjacencies and Lamdas.
// ---------------------------------------------------------------------------
__device__ __forceinline__ void weights4(const float* L1, const float* L2,
                                         const float* L3, const float* L4,
                                         int n, int f,
                                         float& w1, float& w2, float& w3, float& w4) {
    float l1 = L1[n * FDIM + f];
    float l2 = L2[n * FDIM + f];
    float l3 = L3[n * FDIM + f];
    float l4 = L4[n * FDIM + f];
    float m1 = 1.0f - l1;
    float m2 = 1.0f - l2;
    float m3 = 1.0f - l3;
    w1 = l1;
    w2 = m1 * l2;
    w3 = m1 * m2 * l3;
    w4 = m1 * m2 * m3 * l4;
}

__launch_bounds__(256)
__global__ void precompute_kernel(const float* __restrict__ adj,
                                  const float* __restrict__ e1,
                                  const float* __restrict__ e2,
                                  const float* __restrict__ e3,
                                  const float* __restrict__ L1,
                                  const float* __restrict__ L2,
                                  const float* __restrict__ L3,
                                  const float* __restrict__ L4,
                                  float* __restrict__ T,   // [NJ][NJ][FDIM]
                                  float* __restrict__ D)   // [NJ][FDIM]
{
    const int gid    = blockIdx.x * blockDim.x + threadIdx.x;
    const int stride = gridDim.x * blockDim.x;

    // T[n][j][f] = sum_k w_k[n,f] * off_k[n,j]   (off has zero diagonal)
    const int totalT = NJ * NJ * FDIM;
    for (int i = gid; i < totalT; i += stride) {
        int f  = i & (FDIM - 1);
        int nj = i >> 7;
        int n  = nj / NJ;
        int j  = nj - n * NJ;
        float w1, w2, w3, w4;
        weights4(L1, L2, L3, L4, n, f, w1, w2, w3, w4);
        float t = 0.0f;
        if (n != j) {
            float o1 = 0.5f * (adj[n * NJ + j] + adj[j * NJ + n]);
            float o2 = 0.5f * (e1 [n * NJ + j] + e1 [j * NJ + n]);
            float o3 = 0.5f * (e2 [n * NJ + j] + e2 [j * NJ + n]);
            float o4 = 0.5f * (e3 [n * NJ + j] + e3 [j * NJ + n]);
            t = w1 * o1 + w2 * o2 + w3 * o3 + w4 * o4;
        }
        T[i] = t;
    }

    // D[n][f] = sum_k w_k[n,f] * diag_k[n]   (sym diag == original diag)
    const int totalD = NJ * FDIM;
    for (int i = gid; i < totalD; i += stride) {
        int f = i & (FDIM - 1);
        int n = i >> 7;
        float w1, w2, w3, w4;
        weights4(L1, L2, L3, L4, n, f, w1, w2, w3, w4);
        float d1 = adj[n * NJ + n];
        float d2 = e1 [n * NJ + n];
        float d3 = e2 [n * NJ + n];
        float d4 = e3 [n * NJ + n];
        D[i] = w1 * d1 + w2 * d2 + w3 * d3 + w4 * d4;
    }
}

// ---------------------------------------------------------------------------
// Fused main kernel: WMMA fp32 GEMM (h0,h1 into LDS) + graph aggregation.
// LDS: H0 139264B + H1 139264B + T-slice 36992B = 315520B  (<= 320KB/WGP)
// ---------------------------------------------------------------------------
__launch_bounds__(THREADS, 1)
__global__ void fused_graphconv_kernel(const float* __restrict__ X,    // [B*NJ, FDIM]
                                       const float* __restrict__ W,    // [2, FDIM, FDIM]
                                       const float* __restrict__ T,    // [NJ, NJ, FDIM]
                                       const float* __restrict__ D,    // [NJ, FDIM]
                                       const float* __restrict__ bias, // [FDIM]
                                       float* __restrict__ out)        // [B*NJ, FDIM]
{
    __shared__ float lds_h0[ROWS * FDIM];
    __shared__ float lds_h1[ROWS * FDIM];
    __shared__ float lds_T [NJ * NJ * 32];

    const int tid   = threadIdx.x;
    const int wave  = tid >> 5;
    const int lane  = tid & 31;
    const int lhalf = lane >> 4;   // 0: lanes 0-15, 1: lanes 16-31
    const int lmod  = lane & 15;

    const float* W0 = W;
    const float* W1 = W + FDIM * FDIM;
    const int c0      = wave * 16;                 // this wave's column tile
    const int rowBase = blockIdx.x * ROWS;         // first global row

    // ---------------- Stage A: H0 = X@W0, H1 = X@W1 via V_WMMA_F32_16X16X4 --
    // B-fragment layout (4x16 fp32): lane L, vgpr v -> B[k = 2*(L/16)+v][c0 + L%16]
    v2f b0[32], b1[32];
#pragma unroll
    for (int kc = 0; kc < 32; ++kc) {
        int k0 = kc * 4 + 2 * lhalf;
        v2f t0, t1;
        t0.x = W0[(k0    ) * FDIM + c0 + lmod];
        t0.y = W0[(k0 + 1) * FDIM + c0 + lmod];
        t1.x = W1[(k0    ) * FDIM + c0 + lmod];
        t1.y = W1[(k0 + 1) * FDIM + c0 + lmod];
        b0[kc] = t0;
        b1[kc] = t1;
    }

    for (int mt = 0; mt < NJ; ++mt) {   // 17 m-tiles of 16 rows
        // A-fragment layout (16x4 fp32): lane L, vgpr v -> A[L%16][2*(L/16)+v]
        const float* xrow = X + (rowBase + mt * 16 + lmod) * FDIM + 2 * lhalf;
        v8f acc0 = {};
        v8f acc1 = {};
#pragma unroll
        for (int kc = 0; kc < 32; ++kc) {
            v2f a = *(const v2f*)(xrow + kc * 4);
            acc0 = __builtin_amdgcn_wmma_f32_16x16x4_f32(
                false, a, false, b0[kc], (short)0, acc0, false, false);
            acc1 = __builtin_amdgcn_wmma_f32_16x16x4_f32(
                false, a, false, b1[kc], (short)0, acc1, false, false);
        }
        // C/D layout: vgpr i -> row (mt*16 + i + 8*lhalf), col (c0 + lmod)
#pragma unroll
        for (int i = 0; i < 8; ++i) {
            int r = mt * 16 + i + 8 * lhalf;
            lds_h0[r * FDIM + c0 + lmod] = acc0[i];
            lds_h1[r * FDIM + c0 + lmod] = acc1[i];
        }
    }

    // ---------------- Stage B: out = D.*h0 + T (x)_j h1 + bias ------------
    // Each wave handles 2 local batches; lane <-> feature within f-group.
    const int bA = wave * 2;
    const int bB = wave * 2 + 1;

    for (int q = 0; q < 4; ++q) {            // f-groups of 32
        __syncthreads();                     // H ready / previous T-slice consumed
        for (int idx = tid; idx < NJ * NJ * 32; idx += THREADS) {
            int f  = idx & 31;
            int nj = idx >> 5;
            lds_T[idx] = T[nj * FDIM + q * 32 + f];
        }
        __syncthreads();

        const int f = q * 32 + lane;
        const float bval = bias[f];

        float Dreg[NJ];
#pragma unroll
        for (int n = 0; n < NJ; ++n) Dreg[n] = D[n * FDIM + f];

        float h1a[NJ], h1b[NJ];
#pragma unroll
        for (int j = 0; j < NJ; ++j) {
            h1a[j] = lds_h1[(bA * NJ + j) * FDIM + f];
            h1b[j] = lds_h1[(bB * NJ + j) * FDIM + f];
        }

#pragma unroll
        for (int n = 0; n < NJ; ++n) {
            float acca = 0.0f, accb = 0.0f;
#pragma unroll
            for (int j = 0; j < NJ; ++j) {
                float t = lds_T[(n * NJ + j) * 32 + lane];
                acca = fmaf(t, h1a[j], acca);
                accb = fmaf(t, h1b[j], accb);
            }
            float h0a = lds_h0[(bA * NJ + n) * FDIM + f];
            float h0b = lds_h0[(bB * NJ + n) * FDIM + f];
            int ra = (rowBase + bA * NJ + n) * FDIM + f;
            int rb = (rowBase + bB * NJ + n) * FDIM + f;
            out[ra] = fmaf(Dreg[n], h0a, acca) + bval;
            out[rb] = fmaf(Dreg[n], h0b, accb) + bval;
        }
    }
}

// ---------------------------------------------------------------------------
// Launch
// ---------------------------------------------------------------------------
extern "C" void kernel_launch(void* const* d_in, const int* in_sizes, int n_in,
                              void* d_out, int out_size, void* d_ws, size_t ws_size,
                              hipStream_t stream)
{
    const float* x    = (const float*)d_in[0];
    const float* W    = (const float*)d_in[1];
    const float* adj  = (const float*)d_in[2];
    const float* e1   = (const float*)d_in[3];
    const float* e2   = (const float*)d_in[4];
    const float* e3   = (const float*)d_in[5];
    const float* L1   = (const float*)d_in[6];
    const float* L2   = (const float*)d_in[7];
    const float* L3   = (const float*)d_in[8];
    const float* L4   = (const float*)d_in[9];
    const float* bias = (const float*)d_in[10];
    float* out = (float*)d_out;

    float* T = (float*)d_ws;                 // NJ*NJ*FDIM floats
    float* D = T + NJ * NJ * FDIM;           // NJ*FDIM floats

    const int Bbatch = in_sizes[0] / (NJ * FDIM);   // 16384
    const int nblocks = Bbatch / TB;                // 1024

    precompute_kernel<<<16, THREADS, 0, stream>>>(adj, e1, e2, e3,
                                                  L1, L2, L3, L4, T, D);
    fused_graphconv_kernel<<<nblocks, THREADS, 0, stream>>>(x, W, T, D, bias, out);
}